// EMDLoss_86225763435419
// MI455X (gfx1250) — compile-verified
//
#include <hip/hip_runtime.h>
#include <hip/hip_bf16.h>
#include <math.h>

typedef __attribute__((ext_vector_type(2))) float v2f;
typedef __attribute__((ext_vector_type(8))) float v8f;

#define BATCH   8
#define MPTS    4096
#define NPTS    4096
#define KCHUNK  256
#define GCHUNKS (NPTS / KCHUNK)     // 16
#define TILES_PER_CHUNK (KCHUNK/16) // 16
#define WAVES_PER_BLOCK 8

// -10/T * log2(e), for exp(-d/T) = exp2(d * NEG10_LOG2E)
#define NEG10_LOG2E (-14.426950408889634f)

// One wave per (batch, 16-row pred tile). For each 16x16 (pred x gt) tile:
//   A row  = (x, y, z, 1)
//   B col  = (-2x', -2y', -2z', ||g||^2)
//   D = A*B + 0  ->  D[m][n] = ||g_n||^2 - 2<p_m, g_n>
// sq = D + ||p_m||^2 ; d = sqrt(max(sq,1e-12)) ; e = exp(-d/0.1)
// softmin per 256-wide gt chunk: acc += (sum e*d)/(sum e), chunk sums reduced
// across the 16 lanes holding one row via shfl_xor butterflies (wave32 halves).
//
// sq >= 1e-12 (no denorm/zero/inf) so the raw v_sqrt_f32 / v_rcp_f32 /
// v_exp_f32 hardware ops are used directly instead of the precise libm
// expansions — this kernel is TRANS-pipe bound, not memory bound (inputs are
// 786 KB total vs 192 MB L2).
__global__ __launch_bounds__(256)
void emd_tile_kernel(const float* __restrict__ pred,
                     const float* __restrict__ gt,
                     float* __restrict__ partial) {
    const int wid  = threadIdx.x >> 5;
    const int lane = threadIdx.x & 31;
    const int gw   = blockIdx.x * WAVES_PER_BLOCK + wid;   // 0..2047
    const int b    = gw >> 8;                              // 256 tiles / batch
    const int m0   = (gw & 255) << 4;

    const int h = lane >> 4;   // lane half: selects K components (2h, 2h+1)
    const int l = lane & 15;

    const float* __restrict__ predB = pred + (size_t)b * MPTS * 3;
    const float* __restrict__ gtB   = gt   + (size_t)b * NPTS * 3;

    // ---- A fragment (16x4 f32): row M=l, K comps {2h, 2h+1}; K=3 padded w/ 1.0
    const float px = predB[(m0 + l) * 3 + 0];
    const float py = predB[(m0 + l) * 3 + 1];
    const float pz = predB[(m0 + l) * 3 + 2];
    v2f afrag;
    afrag.x = h ? pz   : px;
    afrag.y = h ? 1.0f : py;

    // ---- per-lane ||p||^2 for the 8 rows this lane's half owns in C/D
    float pn[8];
#pragma unroll
    for (int v = 0; v < 8; ++v) {
        const float* p = predB + (size_t)(m0 + 8 * h + v) * 3;
        const float x = p[0], y = p[1], z = p[2];
        pn[v] = x * x + y * y + z * z;
    }

    float acc[8];
#pragma unroll
    for (int v = 0; v < 8; ++v) acc[v] = 0.0f;

    for (int g = 0; g < GCHUNKS; ++g) {
        float se[8], swd[8];
#pragma unroll
        for (int v = 0; v < 8; ++v) { se[v] = 0.0f; swd[v] = 0.0f; }

        for (int t = 0; t < TILES_PER_CHUNK; ++t) {
            const int n = g * KCHUNK + t * 16 + l;
            const float* q = gtB + (size_t)n * 3;
            if (t + 1 < TILES_PER_CHUNK)
                __builtin_prefetch(q + 48, 0, 3);   // next 16-point gt tile
            const float gx = q[0], gy = q[1], gz = q[2];
            const float gn = gx * gx + gy * gy + gz * gz;

            // B fragment (4x16 f32): col N=l, K rows {2h, 2h+1}
            v2f bfrag;
            bfrag.x = h ? (-2.0f * gz) : (-2.0f * gx);
            bfrag.y = h ? gn           : (-2.0f * gy);

            v8f c = {};
            c = __builtin_amdgcn_wmma_f32_16x16x4_f32(
                    /*neg_a=*/false, afrag, /*neg_b=*/false, bfrag,
                    /*c_mod=*/(short)0, c, /*reuse_a=*/false, /*reuse_b=*/false);

#pragma unroll
            for (int v = 0; v < 8; ++v) {
                float sq = fmaxf(c[v] + pn[v], 1e-12f);
                const float d = __builtin_amdgcn_sqrtf(sq);           // v_sqrt_f32
                const float e = __builtin_amdgcn_exp2f(d * NEG10_LOG2E); // v_exp_f32
                se[v]  += e;
                swd[v] += e * d;
            }
        }

        // per-row softmin = (sum e*d)/(sum e): reduce over the 16 lanes of a
        // half (xor offsets 1,2,4,8 never cross the half boundary)
#pragma unroll
        for (int v = 0; v < 8; ++v) {
            float s = se[v], w = swd[v];
#pragma unroll
            for (int off = 1; off < 16; off <<= 1) {
                s += __shfl_xor(s, off, 32);
                w += __shfl_xor(w, off, 32);
            }
            // s in (0, 256], strictly positive: raw v_rcp_f32 is safe
            acc[v] += w * __builtin_amdgcn_rcpf(s);
        }
    }

    // Sum this wave's 16 pred rows: 8 rows per half (replicated), + other half.
    float part = 0.0f;
#pragma unroll
    for (int v = 0; v < 8; ++v) part += acc[v];
    part += __shfl_xor(part, 16, 32);
    if (lane == 0) partial[gw] = part;
}

// Deterministic single-block tree reduction of the 2048 per-wave partials.
__global__ __launch_bounds__(256)
void emd_reduce_kernel(const float* __restrict__ partial, float* __restrict__ out,
                       int n) {
    __shared__ float smem[256];
    float s = 0.0f;
    for (int i = threadIdx.x; i < n; i += 256) s += partial[i];
    smem[threadIdx.x] = s;
    __syncthreads();
    for (int stride = 128; stride > 0; stride >>= 1) {
        if ((int)threadIdx.x < stride) smem[threadIdx.x] += smem[threadIdx.x + stride];
        __syncthreads();
    }
    if (threadIdx.x == 0)
        out[0] = smem[0] * (1.0f / ((float)BATCH * (float)MPTS));  // mean, w=1
}

extern "C" void kernel_launch(void* const* d_in, const int* in_sizes, int n_in,
                              void* d_out, int out_size, void* d_ws, size_t ws_size,
                              hipStream_t stream) {
    (void)in_sizes; (void)n_in; (void)out_size; (void)ws_size;
    const float* pred = (const float*)d_in[0];
    const float* gt   = (const float*)d_in[1];
    float* out        = (float*)d_out;
    float* partial    = (float*)d_ws;                    // 2048 floats = 8 KB

    const int n_waves = BATCH * (MPTS / 16);             // 2048
    dim3 grid(n_waves / WAVES_PER_BLOCK);                // 256 blocks x 8 waves
    emd_tile_kernel<<<grid, 256, 0, stream>>>(pred, gt, partial);
    emd_reduce_kernel<<<1, 256, 0, stream>>>(partial, out, n_waves);
}